// ConvBlock_84361747628702
// MI455X (gfx1250) — compile-verified
//
#include <hip/hip_runtime.h>
#include <math.h>

#define INC    256
#define OUTC   256
#define NHEADS 8
#define HEADC  32
#define BNECK  64
#define NEG_SLOPE 0.2f
#define BN_EPS 1e-5f

typedef float v2f __attribute__((ext_vector_type(2)));
typedef float v8f __attribute__((ext_vector_type(8)));

// ---------------------------------------------------------------------------
// fp32 WMMA GEMM:  C[M,NC] = A[M,K] @ B[K,NC] (+ bias[NC])
// One 16x16 tile per wave, 8 waves per block, K stepped by 4 per wmma.
// ---------------------------------------------------------------------------
__global__ void gemm16x16x4(const float* __restrict__ A,
                            const float* __restrict__ B,
                            const float* __restrict__ bias,
                            float* __restrict__ C,
                            int M, int K, int NC)
{
    const int lane  = threadIdx.x & 31;
    const int wave  = threadIdx.x >> 5;
    const int tileN = blockIdx.x << 4;
    const int tileM = (blockIdx.y * 8 + wave) << 4;
    if (tileM >= M) return;                 // wave-uniform: EXEC stays all-1s

    const int half = lane >> 4;             // 0: K pair {0,1}; 1: K pair {2,3}
    const int l16  = lane & 15;

    v8f acc = {};
    const float* ap = A + (size_t)(tileM + l16) * K + (half << 1);
    const float* bp = B + (size_t)(half << 1) * NC + tileN + l16;

    for (int k = 0; k < K; k += 4) {
        v2f a, b;
        a.x = ap[k];
        a.y = ap[k + 1];
        b.x = bp[(size_t)k * NC];
        b.y = bp[(size_t)(k + 1) * NC];
        acc = __builtin_amdgcn_wmma_f32_16x16x4_f32(
                  false, a, false, b, (short)0, acc, false, false);
    }

    const float bv = bias ? bias[tileN + l16] : 0.0f;
    float* cp = C + (size_t)(tileM + (half << 3)) * NC + tileN + l16;
#pragma unroll
    for (int r = 0; r < 8; ++r)
        cp[(size_t)r * NC] = acc[r] + bv;
}

// ---------------------------------------------------------------------------
// Attention scores: a_src[n,h] = <xh[n,h,:], att_src[h,:]>, same for dst
// ---------------------------------------------------------------------------
__global__ void att_score_kernel(const float* __restrict__ xh,
                                 const float* __restrict__ att_src,
                                 const float* __restrict__ att_dst,
                                 float* __restrict__ a_src,
                                 float* __restrict__ a_dst, int N)
{
    int i = blockIdx.x * blockDim.x + threadIdx.x;
    if (i >= N * NHEADS) return;
    int n = i >> 3, h = i & 7;
    const float* xr = xh + (size_t)n * OUTC + h * HEADC;
    const float* as = att_src + h * HEADC;
    const float* ad = att_dst + h * HEADC;
    float s = 0.f, d = 0.f;
#pragma unroll
    for (int c = 0; c < HEADC; ++c) { float v = xr[c]; s += v * as[c]; d += v * ad[c]; }
    a_src[i] = s;
    a_dst[i] = d;
}

__global__ void fill_kernel(float* __restrict__ p, float v, int n)
{
    int i = blockIdx.x * blockDim.x + threadIdx.x;
    if (i < n) p[i] = v;
}

__device__ __forceinline__ void atomicMaxFloat(float* addr, float val)
{
    if (val >= 0.f) atomicMax((int*)addr, __float_as_int(val));
    else            atomicMin((unsigned int*)addr, __float_as_uint(val));
}

__device__ __forceinline__ float leaky(float v) { return v > 0.f ? v : NEG_SLOPE * v; }

// pass 1: segment max of leaky_relu(a_src[src]+a_dst[dst]) into m[dst,h]
__global__ void edge_max_kernel(const long long* __restrict__ src,
                                const long long* __restrict__ dst,
                                const float* __restrict__ a_src,
                                const float* __restrict__ a_dst,
                                float* __restrict__ m, int E)
{
    int i = blockIdx.x * blockDim.x + threadIdx.x;
    if (i >= E * NHEADS) return;
    int e = i >> 3, h = i & 7;
    int s = (int)src[e], d = (int)dst[e];
    float v = leaky(a_src[s * NHEADS + h] + a_dst[d * NHEADS + h]);
    atomicMaxFloat(&m[d * NHEADS + h], v);
}

// replace -inf (nodes with no in-edges) by 0
__global__ void fix_m_kernel(float* __restrict__ m, int n)
{
    int i = blockIdx.x * blockDim.x + threadIdx.x;
    if (i >= n) return;
    if (m[i] < -3.0e38f) m[i] = 0.f;
}

// pass 2: denom[dst,h] += exp(e - m[dst,h])
__global__ void edge_expsum_kernel(const long long* __restrict__ src,
                                   const long long* __restrict__ dst,
                                   const float* __restrict__ a_src,
                                   const float* __restrict__ a_dst,
                                   const float* __restrict__ m,
                                   float* __restrict__ denom, int E)
{
    int i = blockIdx.x * blockDim.x + threadIdx.x;
    if (i >= E * NHEADS) return;
    int e = i >> 3, h = i & 7;
    int s = (int)src[e], d = (int)dst[e];
    float v = leaky(a_src[s * NHEADS + h] + a_dst[d * NHEADS + h]);
    atomicAdd(&denom[d * NHEADS + h], __expf(v - m[d * NHEADS + h]));
}

// pass 3: out[dst,c] += xh[src,c] * alpha   (one 256-thread block per edge)
__global__ void edge_message_kernel(const long long* __restrict__ src,
                                    const long long* __restrict__ dst,
                                    const float* __restrict__ a_src,
                                    const float* __restrict__ a_dst,
                                    const float* __restrict__ m,
                                    const float* __restrict__ denom,
                                    const float* __restrict__ xh,
                                    float* __restrict__ out, int E)
{
    int e = blockIdx.x;
    int c = threadIdx.x;          // 0..255
    int h = c >> 5;               // head = c / 32
    int s = (int)src[e], d = (int)dst[e];
    float v  = leaky(a_src[s * NHEADS + h] + a_dst[d * NHEADS + h]);
    float ex = __expf(v - m[d * NHEADS + h]);
    float alpha = ex / (denom[d * NHEADS + h] + 1e-16f);
    float val = xh[(size_t)s * OUTC + c] * alpha;
    atomicAdd(&out[(size_t)d * OUTC + c], val);
}

// ---------------------------------------------------------------------------
// column statistics: stats[0..C-1] += sum, stats[C..2C-1] += sumsq
// ---------------------------------------------------------------------------
__global__ void colstats_kernel(const float* __restrict__ X,
                                float* __restrict__ stats, int N, int C)
{
    int t = threadIdx.x;
    int rowsPerBlk = blockDim.x / C;
    int c = t % C;
    int r = blockIdx.x * rowsPerBlk + t / C;
    int stride = gridDim.x * rowsPerBlk;
    float s = 0.f, ss = 0.f;
    for (; r < N; r += stride) {
        float v = X[(size_t)r * C + c];
        s += v; ss += v * v;
    }
    atomicAdd(&stats[c], s);
    atomicAdd(&stats[C + c], ss);
}

// stats -> {mean, rstd}
__global__ void finalize_stats_kernel(float* __restrict__ stats, int N, int C)
{
    int c = blockIdx.x * blockDim.x + threadIdx.x;
    if (c >= C) return;
    float mean = stats[c] / (float)N;
    float var  = stats[C + c] / (float)N - mean * mean;
    stats[c]     = mean;
    stats[C + c] = rsqrtf(fmaxf(var, 0.f) + BN_EPS);
}

__device__ __forceinline__ float elu(float v) { return v > 0.f ? v : __expf(v) - 1.f; }

// Y = elu(bn(X))
__global__ void bn_elu_kernel(const float* __restrict__ X,
                              const float* __restrict__ stats,
                              const float* __restrict__ gamma,
                              const float* __restrict__ beta,
                              float* __restrict__ Y, int N, int C)
{
    int i = blockIdx.x * blockDim.x + threadIdx.x;
    if (i >= N * C) return;
    int c = i % C;
    float v = (X[i] - stats[c]) * stats[C + c] * gamma[c] + beta[c];
    Y[i] = elu(v);
}

// out = h + elu(bn(upre)) + x
__global__ void final_kernel(const float* __restrict__ h,
                             const float* __restrict__ upre,
                             const float* __restrict__ stats,
                             const float* __restrict__ gamma,
                             const float* __restrict__ beta,
                             const float* __restrict__ x,
                             float* __restrict__ out, int N, int C)
{
    int i = blockIdx.x * blockDim.x + threadIdx.x;
    if (i >= N * C) return;
    int c = i % C;
    float v = (upre[i] - stats[c]) * stats[C + c] * gamma[c] + beta[c];
    out[i] = h[i] + elu(v) + x[i];
}

// ---------------------------------------------------------------------------
extern "C" void kernel_launch(void* const* d_in, const int* in_sizes, int n_in,
                              void* d_out, int out_size, void* d_ws, size_t ws_size,
                              hipStream_t stream)
{
    const float*     x         = (const float*)d_in[0];
    const long long* ei        = (const long long*)d_in[1];   // int64 [2,E]
    const float*     W_lin     = (const float*)d_in[2];
    const float*     att_src   = (const float*)d_in[3];
    const float*     att_dst   = (const float*)d_in[4];
    const float*     conv_bias = (const float*)d_in[5];
    const float*     W_res     = (const float*)d_in[6];
    const float*     g_norm    = (const float*)d_in[7];
    const float*     b_norm    = (const float*)d_in[8];
    const float*     W_down    = (const float*)d_in[9];
    const float*     b_down    = (const float*)d_in[10];
    const float*     g_down    = (const float*)d_in[11];
    const float*     bb_down   = (const float*)d_in[12];
    const float*     W_up      = (const float*)d_in[13];
    const float*     b_up      = (const float*)d_in[14];
    const float*     g_up      = (const float*)d_in[15];
    const float*     bb_up     = (const float*)d_in[16];

    const int N = in_sizes[0] / INC;   // 50000
    const int E = in_sizes[1] / 2;     // 800000
    const long long* srcIdx = ei;
    const long long* dstIdx = ei + E;

    // workspace layout (floats)
    float* ws    = (float*)d_ws;
    float* xh    = ws;  ws += (size_t)N * OUTC;    // x @ W_lin
    float* outb  = ws;  ws += (size_t)N * OUTC;    // GAT output accumulator
    float* hbuf  = ws;  ws += (size_t)N * OUTC;    // h = elu(bn(out))
    float* dpre  = ws;  ws += (size_t)N * BNECK;
    float* dbuf  = ws;  ws += (size_t)N * BNECK;
    float* upre  = ws;  ws += (size_t)N * OUTC;
    float* asrc  = ws;  ws += (size_t)N * NHEADS;
    float* adst  = ws;  ws += (size_t)N * NHEADS;
    float* mbuf  = ws;  ws += (size_t)N * NHEADS;
    float* denom = ws;  ws += (size_t)N * NHEADS;
    float* st1   = ws;  ws += 2 * OUTC;
    float* st2   = ws;  ws += 2 * BNECK;
    float* st3   = ws;  ws += 2 * OUTC;

    const dim3 blk(256);
    const int rowBlocks = (N / 16 + 7) / 8;        // 8 waves (tiles) per block

    // --- GEMMs: xh = x@W_lin ; outb = x@W_res + conv_bias (scatter target) ---
    {
        dim3 g(OUTC / 16, rowBlocks);
        gemm16x16x4<<<g, blk, 0, stream>>>(x, W_lin, nullptr,   xh,   N, INC, OUTC);
        gemm16x16x4<<<g, blk, 0, stream>>>(x, W_res, conv_bias, outb, N, INC, OUTC);
    }

    // --- attention scores ---
    att_score_kernel<<<(N * NHEADS + 255) / 256, blk, 0, stream>>>(
        xh, att_src, att_dst, asrc, adst, N);

    // --- segment softmax over destination nodes ---
    fill_kernel<<<(N * NHEADS + 255) / 256, blk, 0, stream>>>(mbuf, -INFINITY, N * NHEADS);
    hipMemsetAsync(denom, 0, (size_t)N * NHEADS * sizeof(float), stream);

    edge_max_kernel<<<(E * NHEADS + 255) / 256, blk, 0, stream>>>(
        srcIdx, dstIdx, asrc, adst, mbuf, E);
    fix_m_kernel<<<(N * NHEADS + 255) / 256, blk, 0, stream>>>(mbuf, N * NHEADS);
    edge_expsum_kernel<<<(E * NHEADS + 255) / 256, blk, 0, stream>>>(
        srcIdx, dstIdx, asrc, adst, mbuf, denom, E);

    // --- message aggregation: outb[dst] += xh[src] * alpha ---
    edge_message_kernel<<<E, blk, 0, stream>>>(
        srcIdx, dstIdx, asrc, adst, mbuf, denom, xh, outb, E);

    // --- BN1 + ELU -> h ---
    hipMemsetAsync(st1, 0, 2 * OUTC * sizeof(float), stream);
    colstats_kernel<<<512, blk, 0, stream>>>(outb, st1, N, OUTC);
    finalize_stats_kernel<<<1, OUTC, 0, stream>>>(st1, N, OUTC);
    bn_elu_kernel<<<(N * OUTC + 255) / 256, blk, 0, stream>>>(
        outb, st1, g_norm, b_norm, hbuf, N, OUTC);

    // --- bottleneck down: dpre = h @ W_down + b_down; BN2 + ELU -> d ---
    {
        dim3 g(BNECK / 16, rowBlocks);
        gemm16x16x4<<<g, blk, 0, stream>>>(hbuf, W_down, b_down, dpre, N, OUTC, BNECK);
    }
    hipMemsetAsync(st2, 0, 2 * BNECK * sizeof(float), stream);
    colstats_kernel<<<512, blk, 0, stream>>>(dpre, st2, N, BNECK);
    finalize_stats_kernel<<<1, BNECK, 0, stream>>>(st2, N, BNECK);
    bn_elu_kernel<<<(N * BNECK + 255) / 256, blk, 0, stream>>>(
        dpre, st2, g_down, bb_down, dbuf, N, BNECK);

    // --- bottleneck up: upre = d @ W_up + b_up; BN3 ---
    {
        dim3 g(OUTC / 16, rowBlocks);
        gemm16x16x4<<<g, blk, 0, stream>>>(dbuf, W_up, b_up, upre, N, BNECK, OUTC);
    }
    hipMemsetAsync(st3, 0, 2 * OUTC * sizeof(float), stream);
    colstats_kernel<<<512, blk, 0, stream>>>(upre, st3, N, OUTC);
    finalize_stats_kernel<<<1, OUTC, 0, stream>>>(st3, N, OUTC);

    // --- out = h + elu(bn(upre)) + x ---
    final_kernel<<<(N * OUTC + 255) / 256, blk, 0, stream>>>(
        hbuf, upre, st3, g_up, bb_up, x, (float*)d_out, N, OUTC);
}